// ExternalKnowledge_14834817040489
// MI455X (gfx1250) — compile-verified
//
#include <hip/hip_runtime.h>
#include <math.h>

typedef __attribute__((ext_vector_type(2))) float v2f;
typedef __attribute__((ext_vector_type(4))) float v4f;
typedef __attribute__((ext_vector_type(8))) float v8f;

#define B_ 8
#define M_ 8192
#define D_ 256
#define MAX_HOP_ 3
#define CHUNKS 64
#define ROWS_PER_CHUNK (M_ / CHUNKS) /* 128 */

// ---------------- init: u = query_vector ----------------
__global__ void k_init_u(const float* __restrict__ q, float* __restrict__ u) {
    int i = blockIdx.x * blockDim.x + threadIdx.x;
    if (i < B_ * D_) u[i] = q[i];
}

// ---------------- logits via V_WMMA_F32_16X16X4_F32 ----------------
// grid: (M_/128, B_), block: 256 threads (8 waves, 16 m-rows per wave).
// A(16x4) = u-chunk broadcast into all 16 rows  (ds_load_b64 from LDS)
// B(4x16) = 16 m-rows x 4 d  (global_load_b64 per lane, rows streamed)
// All D rows identical -> accumulator VGPR0 lanes 0..15 hold the 16 dots.
__global__ void k_logits(const float* __restrict__ mA, const float* __restrict__ gp,
                         const float* __restrict__ u, float* __restrict__ logits) {
    __shared__ float u_lds[D_];
    const int b = blockIdx.y;
    if (threadIdx.x < D_) u_lds[threadIdx.x] = u[b * D_ + threadIdx.x];
    __syncthreads();

    const int wave = threadIdx.x >> 5;
    const int lane = threadIdx.x & 31;
    const int half = lane >> 4;   // 0: K=0,1   1: K=2,3
    const int r    = lane & 15;   // row-in-tile / N index
    const int row  = blockIdx.x * 128 + wave * 16 + r;

    const float* rowp = mA + ((size_t)b * M_ + row) * D_;

    v8f c = {0.f, 0.f, 0.f, 0.f, 0.f, 0.f, 0.f, 0.f};
    #pragma unroll 8
    for (int t = 0; t < D_ / 4; ++t) {
        const int d0 = 4 * t + 2 * half;
        v2f a  = *(const v2f*)(&u_lds[d0]);   // broadcast u chunk (A operand)
        v2f bb = *(const v2f*)(rowp + d0);    // m-row chunk        (B operand)
        c = __builtin_amdgcn_wmma_f32_16x16x4_f32(
                /*neg_a=*/false, a, /*neg_b=*/false, bb,
                /*c_mod=*/(short)0, c, /*reuse_a=*/false, /*reuse_b=*/false);
    }
    if (half == 0) {
        // logits = gp * dot(m_row, u); lanes 0..15 of c[0] hold the 16 dots
        logits[b * M_ + row] = gp[b * M_ + row] * c[0];
    }
}

// ---------------- softmax over m (one workgroup per b) ----------------
__global__ void k_softmax(const float* __restrict__ logits, float* __restrict__ prob) {
    __shared__ float red[1024];
    const int b = blockIdx.x;
    const int tid = threadIdx.x;
    const float* lb = logits + b * M_;

    float mx = -INFINITY;
    for (int i = tid; i < M_; i += 1024) mx = fmaxf(mx, lb[i]);
    red[tid] = mx;
    __syncthreads();
    for (int s = 512; s > 0; s >>= 1) {
        if (tid < s) red[tid] = fmaxf(red[tid], red[tid + s]);
        __syncthreads();
    }
    const float bmax = red[0];
    __syncthreads();

    float sm = 0.f;
    for (int i = tid; i < M_; i += 1024) sm += expf(lb[i] - bmax);
    red[tid] = sm;
    __syncthreads();
    for (int s = 512; s > 0; s >>= 1) {
        if (tid < s) red[tid] += red[tid + s];
        __syncthreads();
    }
    const float inv = 1.0f / red[0];

    for (int i = tid; i < M_; i += 1024)
        prob[b * M_ + i] = expf(lb[i] - bmax) * inv;
}

// ---------------- o_k partial sums (b128 streaming reduction) ----------------
// grid: (CHUNKS, B_), block: 256.
// Thread t handles d-columns [4*(t&63) .. +3] for rows (t>>6) + 4k in the chunk:
// one global_load_b128 per row per thread group -> 512B useful data / VMEM instr.
// Cross-row-group reduction via LDS in a fixed order (deterministic).
__global__ void k_ok_partial(const float* __restrict__ mC, const float* __restrict__ gp,
                             const float* __restrict__ prob, float* __restrict__ partial) {
    __shared__ float w[ROWS_PER_CHUNK];
    __shared__ v4f  red4[256];
    const int b = blockIdx.y, chunk = blockIdx.x, tid = threadIdx.x;
    const int mbase = chunk * ROWS_PER_CHUNK;

    if (tid < ROWS_PER_CHUNK) {
        const int m = mbase + tid;
        w[tid] = prob[b * M_ + m] * gp[b * M_ + m];
    }
    __syncthreads();

    const int cg = tid & 63;  // column group: d = 4*cg .. 4*cg+3
    const int rg = tid >> 6;  // row subgroup 0..3

    const float* base = mC + ((size_t)b * M_ + mbase) * D_ + 4 * cg;
    v4f acc = {0.f, 0.f, 0.f, 0.f};
    #pragma unroll 4
    for (int r2 = rg; r2 < ROWS_PER_CHUNK; r2 += 4) {
        v4f v = *(const v4f*)(base + (size_t)r2 * D_);   // global_load_b128
        const float wv = w[r2];
        acc[0] = fmaf(wv, v[0], acc[0]);
        acc[1] = fmaf(wv, v[1], acc[1]);
        acc[2] = fmaf(wv, v[2], acc[2]);
        acc[3] = fmaf(wv, v[3], acc[3]);
    }
    red4[tid] = acc;
    __syncthreads();

    if (tid < 64) {
        v4f a0 = red4[tid], a1 = red4[tid + 64], a2 = red4[tid + 128], a3 = red4[tid + 192];
        v4f s;
        s[0] = (a0[0] + a1[0]) + (a2[0] + a3[0]);
        s[1] = (a0[1] + a1[1]) + (a2[1] + a3[1]);
        s[2] = (a0[2] + a1[2]) + (a2[2] + a3[2]);
        s[3] = (a0[3] + a1[3]) + (a2[3] + a3[3]);
        *(v4f*)&partial[((size_t)b * CHUNKS + chunk) * D_ + 4 * tid] = s;
    }
}

// ---------------- u += sum of partials (fixed order -> deterministic) ----------------
__global__ void k_u_update(const float* __restrict__ partial, float* __restrict__ u) {
    const int b = blockIdx.x, d = threadIdx.x;
    float acc = 0.f;
    for (int c = 0; c < CHUNKS; ++c)
        acc += partial[((size_t)b * CHUNKS + c) * D_ + d];
    u[b * D_ + d] += acc;
}

extern "C" void kernel_launch(void* const* d_in, const int* in_sizes, int n_in,
                              void* d_out, int out_size, void* d_ws, size_t ws_size,
                              hipStream_t stream) {
    const float* q  = (const float*)d_in[0];  // query_vector (B, D)
    const float* gp = (const float*)d_in[1];  // global_pointer (B, M)
    const float* ms = (const float*)d_in[2];  // m_story (MAX_HOP+1, B, M, D)

    float* out        = (float*)d_out;
    float* prob_out   = out;            // prob_soft  (B, M)  -- tuple element 0
    float* logits_out = out + B_ * M_;  // prob_logits(B, M)  -- tuple element 1

    float* u       = (float*)d_ws;      // B*D floats
    float* partial = u + B_ * D_;       // B*CHUNKS*D floats (512 KB)

    const size_t S = (size_t)B_ * M_ * D_;  // one m_story slice

    k_init_u<<<(B_ * D_ + 255) / 256, 256, 0, stream>>>(q, u);

    for (int hop = 0; hop < MAX_HOP_; ++hop) {
        k_logits<<<dim3(M_ / 128, B_), 256, 0, stream>>>(ms + (size_t)hop * S, gp, u, logits_out);
        k_softmax<<<B_, 1024, 0, stream>>>(logits_out, prob_out);
        k_ok_partial<<<dim3(CHUNKS, B_), 256, 0, stream>>>(ms + (size_t)(hop + 1) * S, gp, prob_out, partial);
        k_u_update<<<B_, D_, 0, stream>>>(partial, u);
    }
    // After the final hop, d_out holds the last hop's prob_soft and prob_logits.
}